// SelfDistillationGraphConvGNN_77326591197423
// MI455X (gfx1250) — compile-verified
//
#include <hip/hip_runtime.h>
#include <hip/hip_bf16.h>

// ---------------- problem constants ----------------
#define N_U   50000
#define N_I   50000
#define FH    128          // F == H == 128
#define NE    800000
#define NG    512
#define NC    16
#define NL    3
#define EPSV  1e-5f

typedef float v2f __attribute__((ext_vector_type(2)));
typedef float v8f __attribute__((ext_vector_type(8)));

__device__ __forceinline__ void atomAddF(float* p, float v) {
    unsafeAtomicAdd(p, v);   // HW global_atomic_add_f32
}

// ---------------- zero fill (grid-stride) ----------------
__global__ void k_zero(float* __restrict__ p, long n) {
    long i = (long)blockIdx.x * blockDim.x + threadIdx.x;
    long s = (long)gridDim.x * blockDim.x;
    for (; i < n; i += s) p[i] = 0.0f;
}

// ---------------- per-graph node counts ----------------
__global__ void k_count(const int* __restrict__ bu, const int* __restrict__ bi,
                        float* __restrict__ cnt) {
    int t = blockIdx.x * blockDim.x + threadIdx.x;
    if (t < N_U)              atomAddF(&cnt[bu[t]], 1.0f);
    else if (t < N_U + N_I)   atomAddF(&cnt[bi[t - N_U]], 1.0f);
}

// ---------------- edge scatter: agg[dst] += x[src] ----------------
__global__ __launch_bounds__(256) void k_scatter(const float* __restrict__ x,
                                                 const int* __restrict__ esrc,
                                                 const int* __restrict__ edst,
                                                 float* __restrict__ agg) {
    long tid = (long)blockIdx.x * blockDim.x + threadIdx.x;
    if (tid >= (long)NE * 32) return;
    int e = (int)(tid >> 5);
    int f = (int)(tid & 31) << 2;
    int s = esrc[e], d = edst[e];
    float4 v = *(const float4*)(x + (long)s * FH + f);
    float* p = agg + (long)d * FH + f;
    atomAddF(p + 0, v.x); atomAddF(p + 1, v.y);
    atomAddF(p + 2, v.z); atomAddF(p + 3, v.w);
}

// ---------------- fused dual-GEMM + bias + ReLU + BN-stats ----------------
// out[n,h] = relu( sum_f agg[n,f]*Wrel[h,f] + sum_f xroot[n,f]*Wroot[h,f] + bias[h] )
// One wave32 -> 16 rows x 128 cols via V_WMMA_F32_16X16X4_F32 (fp32 acc).
__global__ __launch_bounds__(256) void k_gemm_relu_bnstat(
        const float* __restrict__ agg,  const float* __restrict__ xroot,
        const float* __restrict__ Wrel, const float* __restrict__ Wroot,
        const float* __restrict__ bias, float* __restrict__ out,
        float* __restrict__ bnsum, float* __restrict__ bnsq, int nRowTiles)
{
    int wave = blockIdx.x * (blockDim.x >> 5) + (threadIdx.x >> 5);
    if (wave >= nRowTiles) return;                 // wave-uniform: EXEC all-1 inside
    int lane = threadIdx.x & 31;
    int m    = lane & 15;                          // A: row M ; B: col N
    int kh   = (lane >> 4) << 1;                   // K sub-pair 0 or 2
    int n0   = wave << 4;

    v8f acc[8];
    #pragma unroll
    for (int j = 0; j < 8; ++j)
        #pragma unroll
        for (int r = 0; r < 8; ++r) acc[j][r] = 0.0f;

    #pragma unroll
    for (int t = 0; t < 2; ++t) {
        const float* A = t ? xroot : agg;
        const float* W = t ? Wroot : Wrel;
        const float* arow = A + (long)(n0 + m) * FH + kh;
        const float* wrow = W + (long)m * FH + kh;      // row h = j*16 + m handled by j*2048
        for (int k = 0; k < FH; k += 4) {
            v2f a = *(const v2f*)(arow + k);
            #pragma unroll
            for (int j = 0; j < 8; ++j) {
                v2f b = *(const v2f*)(wrow + j * 2048 + k);
                acc[j] = __builtin_amdgcn_wmma_f32_16x16x4_f32(
                             false, a, false, b, (short)0, acc[j], false, false);
            }
        }
    }

    int rowoff = (lane >> 4) << 3;                 // 0 or 8
    #pragma unroll
    for (int j = 0; j < 8; ++j) {
        int col = (j << 4) + m;
        float bv = bias[col];
        float s = 0.0f, q = 0.0f;
        #pragma unroll
        for (int r = 0; r < 8; ++r) {
            float v = acc[j][r] + bv;
            v = v > 0.0f ? v : 0.0f;               // ReLU
            out[(long)(n0 + rowoff + r) * FH + col] = v;
            s += v; q += v * v;
        }
        atomAddF(&bnsum[col], s);
        atomAddF(&bnsq[col],  q);
    }
}

// ---------------- fold BN stats into per-column scale/shift ----------------
// sums layout: [sum_u(128), sq_u(128), sum_i(128), sq_i(128)]
// scsh layout: [scale_u(128), shift_u(128), scale_i(128), shift_i(128)]
__global__ void k_bn_prep(const float* __restrict__ sums,
                          const float* __restrict__ gu, const float* __restrict__ bu,
                          const float* __restrict__ gi, const float* __restrict__ bi,
                          float* __restrict__ scsh) {
    int t = threadIdx.x;            // 0..255
    int col  = t & 127;
    int type = t >> 7;
    const float* s = sums + type * 256;
    float invN = 1.0f / (type ? (float)N_I : (float)N_U);
    float mean = s[col] * invN;
    float var  = s[128 + col] * invN - mean * mean;
    float g = type ? gi[col] : gu[col];
    float b = type ? bi[col] : bu[col];
    float sc = g * rsqrtf(var + EPSV);
    scsh[type * 256 + col]       = sc;
    scsh[type * 256 + 128 + col] = b - mean * sc;
}

// ---------------- apply BN in place + accumulate graph pool ----------------
__global__ __launch_bounds__(256) void k_bn_apply_pool(
        float* __restrict__ x, const int* __restrict__ batch,
        const float* __restrict__ scale, const float* __restrict__ shift,
        float* __restrict__ pooled, int n) {
    long tid = (long)blockIdx.x * blockDim.x + threadIdx.x;
    if (tid >= (long)n * 32) return;
    int node = (int)(tid >> 5);
    int f    = (int)(tid & 31) << 2;
    float* xp = x + (long)node * FH + f;
    float4 v = *(const float4*)xp;
    v.x = v.x * scale[f + 0] + shift[f + 0];
    v.y = v.y * scale[f + 1] + shift[f + 1];
    v.z = v.z * scale[f + 2] + shift[f + 2];
    v.w = v.w * scale[f + 3] + shift[f + 3];
    *(float4*)xp = v;
    float* p = pooled + (long)batch[node] * FH + f;
    atomAddF(p + 0, v.x); atomAddF(p + 1, v.y);
    atomAddF(p + 2, v.z); atomAddF(p + 3, v.w);
}

// ---------------- pooled / max(cnt,1) -> feats[l] ----------------
__global__ void k_feats(const float* __restrict__ pooled, const float* __restrict__ cnt,
                        float* __restrict__ feats) {
    int idx = blockIdx.x * blockDim.x + threadIdx.x;   // < NG*FH
    if (idx >= NG * FH) return;
    float c = cnt[idx >> 7];
    c = c > 1.0f ? c : 1.0f;
    feats[idx] = pooled[idx] / c;
}

// ---------------- per-layer classifier heads ----------------
__global__ void k_logits(const float* __restrict__ feats, const float* __restrict__ fcW,
                         const float* __restrict__ fcb, float* __restrict__ logits) {
    int tid = blockIdx.x * blockDim.x + threadIdx.x;   // < NL*NG*NC
    if (tid >= NL * NG * NC) return;
    int c = tid & (NC - 1);
    int g = (tid >> 4) & (NG - 1);
    int l = tid >> 13;                                  // NG*NC = 8192
    const float4* fr = (const float4*)(feats + ((long)l * NG + g) * FH);
    const float4* wr = (const float4*)(fcW  + ((long)l * NC + c) * FH);
    float s = 0.0f;
    #pragma unroll
    for (int i = 0; i < FH / 4; ++i) {
        float4 a = fr[i], b = wr[i];
        s += a.x * b.x + a.y * b.y + a.z * b.z + a.w * b.w;
    }
    logits[tid] = s + fcb[l * NC + c];
}

// ---------------- workspace layout (float offsets) ----------------
#define OFF_XUA   0L
#define OFF_XUB   (OFF_XUA + (long)N_U * FH)
#define OFF_XIA   (OFF_XUB + (long)N_U * FH)
#define OFF_XIB   (OFF_XIA + (long)N_I * FH)
#define OFF_AGGU  (OFF_XIB + (long)N_I * FH)
#define OFF_AGGI  (OFF_AGGU + (long)N_U * FH)
#define OFF_BNSUM (OFF_AGGI + (long)N_I * FH)   // 512
#define OFF_POOL  (OFF_BNSUM + 512L)            // NG*FH
#define OFF_SCSH  (OFF_POOL + (long)NG * FH)    // 512
#define OFF_CNT   (OFF_SCSH + 512L)             // NG
#define ZERO_LAYER_LEN ((long)N_U * FH + (long)N_I * FH + 512L + (long)NG * FH)

extern "C" void kernel_launch(void* const* d_in, const int* in_sizes, int n_in,
                              void* d_out, int out_size, void* d_ws, size_t ws_size,
                              hipStream_t stream) {
    const float* x_user  = (const float*)d_in[0];
    const float* x_item  = (const float*)d_in[1];
    const int*   e_u2i   = (const int*)d_in[2];    // [2,E]
    const int*   e_i2u   = (const int*)d_in[3];    // [2,E]
    const int*   b_user  = (const int*)d_in[4];
    const int*   b_item  = (const int*)d_in[5];
    const float* Wrel_u2i  = (const float*)d_in[6];
    const float* Wroot_u2i = (const float*)d_in[7];
    const float* bias_u2i  = (const float*)d_in[8];
    const float* Wrel_i2u  = (const float*)d_in[9];
    const float* Wroot_i2u = (const float*)d_in[10];
    const float* bias_i2u  = (const float*)d_in[11];
    const float* bn_g_u = (const float*)d_in[12];
    const float* bn_b_u = (const float*)d_in[13];
    const float* bn_g_i = (const float*)d_in[14];
    const float* bn_b_i = (const float*)d_in[15];
    const float* fc_W   = (const float*)d_in[16];
    const float* fc_b   = (const float*)d_in[17];

    float* ws = (float*)d_ws;
    float* bufU[2] = { ws + OFF_XUA, ws + OFF_XUB };
    float* bufI[2] = { ws + OFF_XIA, ws + OFF_XIB };
    float* aggU  = ws + OFF_AGGU;
    float* aggI  = ws + OFF_AGGI;
    float* bnsum = ws + OFF_BNSUM;      // [su(128), qu(128), si(128), qi(128)]
    float* pool  = ws + OFF_POOL;
    float* scsh  = ws + OFF_SCSH;       // [scu(128), shu(128), sci(128), shi(128)]
    float* cnt   = ws + OFF_CNT;

    float* logits = (float*)d_out;                       // [L,G,C]
    float* feats  = (float*)d_out + (long)NL * NG * NC;  // [L,G,H]

    // per-graph counts (once per call)
    k_zero<<<2, 256, 0, stream>>>(cnt, (long)NG);
    k_count<<<(N_U + N_I + 255) / 256, 256, 0, stream>>>(b_user, b_item, cnt);

    const int scatterBlocks = (int)(((long)NE * 32 + 255) / 256);
    const int rowTiles = N_U / 16;                      // 3125 (exact)
    const int gemmBlocks = (rowTiles + 7) / 8;          // 8 waves / block

    for (int l = 0; l < NL; ++l) {
        const float* xu = (l == 0) ? x_user : bufU[(l + 1) & 1];
        const float* xi = (l == 0) ? x_item : bufI[(l + 1) & 1];
        float* xu_new = bufU[l & 1];
        float* xi_new = bufI[l & 1];

        // clear agg buffers + BN stats + pooled (contiguous region)
        k_zero<<<4096, 256, 0, stream>>>(aggU, ZERO_LAYER_LEN);

        // edge aggregation
        k_scatter<<<scatterBlocks, 256, 0, stream>>>(xu, e_u2i, e_u2i + NE, aggI);
        k_scatter<<<scatterBlocks, 256, 0, stream>>>(xi, e_i2u, e_i2u + NE, aggU);

        // fused dual-GEMM + bias + ReLU + BN stats (WMMA f32 16x16x4)
        k_gemm_relu_bnstat<<<gemmBlocks, 256, 0, stream>>>(
            aggI, xi, Wrel_u2i + (long)l * FH * FH, Wroot_u2i + (long)l * FH * FH,
            bias_u2i + (long)l * FH, xi_new, bnsum + 256, bnsum + 384, rowTiles);
        k_gemm_relu_bnstat<<<gemmBlocks, 256, 0, stream>>>(
            aggU, xu, Wrel_i2u + (long)l * FH * FH, Wroot_i2u + (long)l * FH * FH,
            bias_i2u + (long)l * FH, xu_new, bnsum + 0, bnsum + 128, rowTiles);

        // BN fold + apply + pool
        k_bn_prep<<<1, 256, 0, stream>>>(bnsum, bn_g_u, bn_b_u, bn_g_i, bn_b_i, scsh);
        k_bn_apply_pool<<<(int)(((long)N_U * 32 + 255) / 256), 256, 0, stream>>>(
            xu_new, b_user, scsh + 0,   scsh + 128, pool, N_U);
        k_bn_apply_pool<<<(int)(((long)N_I * 32 + 255) / 256), 256, 0, stream>>>(
            xi_new, b_item, scsh + 256, scsh + 384, pool, N_I);

        // feats[l] = pooled / max(cnt,1)
        k_feats<<<(NG * FH + 255) / 256, 256, 0, stream>>>(pool, cnt, feats + (long)l * NG * FH);
    }

    // classifier heads
    k_logits<<<(NL * NG * NC + 255) / 256, 256, 0, stream>>>(feats, fc_W, fc_b, logits);
}